// heatmap_fit_33277406609539
// MI455X (gfx1250) — compile-verified
//
#include <hip/hip_runtime.h>
#include <stdint.h>

typedef __attribute__((ext_vector_type(2))) float v2f;
typedef __attribute__((ext_vector_type(8))) float v8f;
typedef int v4i __attribute__((vector_size(16)));   // 4 x i32, matches builtin param

#define H_DIM 256
#define W_DIM 256
#define SLICE_ELEMS (H_DIM * W_DIM)          // 65536
#define PANEL_COLS 32                        // k-panel width
#define LDS_ROW_STRIDE 36                    // 32 floats + 16B pad: bank-conflict free, b128 aligned
#define PANEL_FLOATS (16 * LDS_ROW_STRIDE)   // 576 floats per panel
#define NUM_PANELS 8                         // 256 / 32 k-panels per row-tile

// ---- CDNA5 async-copy helpers ---------------------------------------------
__device__ __forceinline__ void async_copy16(float* dst_lds, const float* src_g) {
#if __has_builtin(__builtin_amdgcn_global_load_async_to_lds_b128)
    __builtin_amdgcn_global_load_async_to_lds_b128(
        (__attribute__((address_space(1))) v4i*)(void*)src_g,
        (__attribute__((address_space(3))) v4i*)(void*)dst_lds,
        /*imm offset*/0, /*cpol*/0);
#else
    *(float4*)dst_lds = *(const float4*)src_g;
#endif
}

__device__ __forceinline__ void wait_async0() {
#if __has_builtin(__builtin_amdgcn_s_wait_asynccnt)
    __builtin_amdgcn_s_wait_asynccnt(0);
#else
    asm volatile("s_wait_asynccnt 0" ::: "memory");
#endif
}

__device__ __forceinline__ void wait_async_le4() {
#if __has_builtin(__builtin_amdgcn_s_wait_asynccnt)
    __builtin_amdgcn_s_wait_asynccnt(4);
#else
    asm volatile("s_wait_asynccnt 4" ::: "memory");
#endif
}

__device__ __forceinline__ void wait_async_le8() {
#if __has_builtin(__builtin_amdgcn_s_wait_asynccnt)
    __builtin_amdgcn_s_wait_asynccnt(8);
#else
    asm volatile("s_wait_asynccnt 8" ::: "memory");
#endif
}

__device__ __forceinline__ void wait_ds0() {
#if __has_builtin(__builtin_amdgcn_s_wait_dscnt)
    __builtin_amdgcn_s_wait_dscnt(0);
#else
    asm volatile("s_wait_dscnt 0" ::: "memory");
#endif
}

// ---------------------------------------------------------------------------
// Pass 1: per-slice moments via V_WMMA_F32_16X16X4_F32.
//   D = A(16x4 f32 tile of x) x B(4x16 const: col0 = 1, col1 = k), accumulated
//   over K=256  =>  D[m,0] = rowsum(m), D[m,1] = sum_c x[m,c]*c.
// One block per (b,k) slice; 8 waves; wave w handles row-tiles w and w+8.
// Triple-buffered, depth-2 async b128 global->LDS pipeline (ASYNCcnt).
// B kept as a 2-VGPR running value (opaque to the optimizer) to cap register
// pressure; A-fragments preloaded so DS latency overlaps the WMMA burst.
// ---------------------------------------------------------------------------
__global__ void __launch_bounds__(256)
moments_kernel(const float* __restrict__ x, float* __restrict__ ws) {
    __shared__ float lds[8 * 3 * PANEL_FLOATS];
    __shared__ float parts[8][4];

    const int slice = blockIdx.x;
    const int wave  = threadIdx.x >> 5;
    const int lane  = threadIdx.x & 31;
    const int n     = lane & 15;        // N index (and A's M index)
    const int half  = lane >> 4;        // lane half selects K pair {0,1} vs {2,3}
    const int khalf = half * 2;
    const int mOff  = half * 8;         // C/D: lanes 16-31 hold M = j+8

    float* const wbase = &lds[wave * 3 * PANEL_FLOATS];
    float* bufs[3] = { wbase, wbase + PANEL_FLOATS, wbase + 2 * PANEL_FLOATS };
    const float* src = x + (size_t)slice * SLICE_ELEMS;
    const int mBase = n * LDS_ROW_STRIDE;       // A-fragment row base in panel

    // async-load lane mapping: 4 rows x 8 chunks of 16B per instruction
    const int arow  = lane >> 3;                // 0..3
    const int achnk = (lane & 7) * 4;           // float offset of 16B chunk

    // constant per-lane B data (built once, advanced incrementally)
    const float bx0   = (n == 0) ? 1.0f : ((n == 1) ? (float)khalf       : 0.0f);
    const float by0   = (n == 0) ? 1.0f : ((n == 1) ? (float)(khalf + 1) : 0.0f);
    const float bstep = (n == 1) ? 4.0f : 0.0f; // K advances by 4 per WMMA

    float s0 = 0.f, s1 = 0.f, s2 = 0.f;

    for (int t = 0; t < 2; ++t) {
        const int rowBase = (wave + t * 8) * 16;
        v8f acc = {0.f, 0.f, 0.f, 0.f, 0.f, 0.f, 0.f, 0.f};
        float bx = bx0, by = by0;

        // prime the pipeline: panels 0 and 1
#pragma unroll
        for (int p = 0; p < 2; ++p) {
#pragma unroll
            for (int i = 0; i < 4; ++i) {
                const int row = 4 * i + arow;
                async_copy16(bufs[p] + row * LDS_ROW_STRIDE + achnk,
                             src + (rowBase + row) * W_DIM + p * PANEL_COLS + achnk);
            }
        }

#pragma unroll
        for (int kp = 0; kp < NUM_PANELS; ++kp) {
            // prefetch panel kp+2 (depth-2 lookahead)
            if (kp + 2 < NUM_PANELS) {
                float* nxt = bufs[(kp + 2) % 3];
                const int kbn = (kp + 2) * PANEL_COLS;
#pragma unroll
                for (int i = 0; i < 4; ++i) {
                    const int row = 4 * i + arow;
                    async_copy16(nxt + row * LDS_ROW_STRIDE + achnk,
                                 src + (rowBase + row) * W_DIM + kbn + achnk);
                }
            }
            // async loads complete in order: ensure panel kp landed
            if (kp < NUM_PANELS - 2)       wait_async_le8();
            else if (kp == NUM_PANELS - 2) wait_async_le4();
            else                           wait_async0();

            float* cur = bufs[kp % 3];
            // preload all 8 A-fragments (4 ds_load_2addr_b64 in flight together)
            v2f a[8];
#pragma unroll
            for (int ks = 0; ks < 8; ++ks)
                a[ks] = *(const v2f*)(cur + mBase + ks * 4 + khalf);
            // ---- 8 WMMA k-steps; B advanced by 2 in-loop v_add_f32 ----
#pragma unroll
            for (int ks = 0; ks < 8; ++ks) {
                asm volatile("" : "+v"(bx), "+v"(by));   // keep B a running 2-reg value
                v2f b = {bx, by};
                acc = __builtin_amdgcn_wmma_f32_16x16x4_f32(
                    false, a[ks], false, b, (short)0, acc, false, false);
                bx += bstep;
                by += bstep;
            }
            wait_ds0();   // fragment reads retired before this buffer is reused (WAR)
        }
        // ---- extract: D[m,0]=rowsum, D[m,1]=weighted-col sum ----
#pragma unroll
        for (int j = 0; j < 8; ++j) {
            const float val = acc[j];               // (M=j+mOff, N=n)
            if (n == 0) {
                s0 += val;
                s2 += (float)(rowBase + j + mOff) * val;
            } else if (n == 1) {
                s1 += val;
            }
        }
    }

    // wave reduction (5 xor steps)
#pragma unroll
    for (int m = 16; m >= 1; m >>= 1) {
        s0 += __shfl_xor(s0, m, 32);
        s1 += __shfl_xor(s1, m, 32);
        s2 += __shfl_xor(s2, m, 32);
    }
    if (lane == 0) { parts[wave][0] = s0; parts[wave][1] = s1; parts[wave][2] = s2; }
    __syncthreads();

    if (threadIdx.x == 0) {
        float S0 = 0.f, S1 = 0.f, S2 = 0.f;
        for (int w = 0; w < 8; ++w) { S0 += parts[w][0]; S1 += parts[w][1]; S2 += parts[w][2]; }
        const float mu_x = S1 / S0;     // column-weighted center
        const float mu_y = S2 / S0;     // row-weighted center
        // closed-form grid second moments (H=W=256): mean(c^2)=21717.5, 2*mean(c)=255
        const float var_x = 21717.5f - 255.0f * mu_x + mu_x * mu_x;
        const float var_y = 21717.5f - 255.0f * mu_y + mu_y * mu_y;
        const float cov   = (127.5f - mu_x) * (127.5f - mu_y);
        const float det   = var_x * var_y - cov * cov;
        float* o = ws + (size_t)slice * 8;
        o[0] = mu_x;
        o[1] = mu_y;
        o[2] = var_y / det;     // a
        o[3] = -cov  / det;     // b
        o[4] = var_x / det;     // c
    }
}

// ---------------------------------------------------------------------------
// Pass 2: write-only map. out = 1/(1+q), q = a*v0^2 + 2b*v0*v1 + c*v1^2 with
// the reference's quirk v0 = row - mu_x, v1 = col - mu_y. Never re-reads x.
// grid = (16, slices); 4096 pixels per block, float4 stores.
// ---------------------------------------------------------------------------
__global__ void __launch_bounds__(256)
emit_kernel(const float* __restrict__ ws, float* __restrict__ out) {
    const int slice = blockIdx.y;
    const float* cf = ws + (size_t)slice * 8;
    const float mu_x = cf[0], mu_y = cf[1];
    const float A = cf[2], B = cf[3], C = cf[4];

    float4* out4 = (float4*)out + (size_t)slice * (SLICE_ELEMS / 4);
    const int f4base = blockIdx.x * 1024;   // float4 index within slice

#pragma unroll
    for (int it = 0; it < 4; ++it) {
        const int f4 = f4base + it * 256 + threadIdx.x;
        const int lp = f4 << 2;             // pixel index within slice
        const int r  = lp >> 8;
        const int c0 = lp & 255;
        const float v0  = (float)r - mu_x;
        const float t0  = A * v0 * v0;      // a*v0^2
        const float t1  = 2.0f * B * v0;    // coefficient of v1
        float4 o;
        {
            float v1 = (float)(c0 + 0) - mu_y;
            o.x = 1.0f / (1.0f + t0 + t1 * v1 + C * v1 * v1);
            v1 = (float)(c0 + 1) - mu_y;
            o.y = 1.0f / (1.0f + t0 + t1 * v1 + C * v1 * v1);
            v1 = (float)(c0 + 2) - mu_y;
            o.z = 1.0f / (1.0f + t0 + t1 * v1 + C * v1 * v1);
            v1 = (float)(c0 + 3) - mu_y;
            o.w = 1.0f / (1.0f + t0 + t1 * v1 + C * v1 * v1);
        }
        out4[f4] = o;
    }
}

extern "C" void kernel_launch(void* const* d_in, const int* in_sizes, int n_in,
                              void* d_out, int out_size, void* d_ws, size_t ws_size,
                              hipStream_t stream) {
    const float* x = (const float*)d_in[0];
    float* out = (float*)d_out;
    float* ws  = (float*)d_ws;
    const int slices = in_sizes[0] / SLICE_ELEMS;   // 16*32 = 512

    moments_kernel<<<dim3(slices), dim3(256), 0, stream>>>(x, ws);
    emit_kernel<<<dim3(16, slices), dim3(256), 0, stream>>>(ws, out);
}